// LearnedRouter_11390253269625
// MI455X (gfx1250) — compile-verified
//
#include <hip/hip_runtime.h>
#include <math.h>
#include <stdint.h>

typedef __attribute__((ext_vector_type(2))) float v2f;
typedef __attribute__((ext_vector_type(8))) float v8f;

#define B_ 4
#define S_ 4096
#define D_ 1024
#define N_ 256
#define K_ 8
#define NEGV (-1e9f)
#define MIN_T 0.5f
#define SCALE 0.03125f /* 1/sqrt(1024) */

#define TPAD 4   /* token-tile row pad: (4*row + k) % 64 is a bank permutation */
#define SPAD 4   /* score-tile row pad: rows m and m+8 land 32 banks apart     */

// ---------------------------------------------------------------------------
// Kernel 1: Mt[d][n] = sum_e desc[n][e] * Wq[e][d]   (fused router matrix,
// stored transposed so the main kernel's B-fragment loads are coalesced).
// One wave per 16x16 (n,d) tile, K-loop over e in steps of 4 (f32 WMMA).
// ---------------------------------------------------------------------------
__global__ __launch_bounds__(32)
void fuse_router_kernel(const float* __restrict__ desc,
                        const float* __restrict__ Wq,
                        float* __restrict__ Mt) {
    const int d0   = blockIdx.x * 16;
    const int n0   = blockIdx.y * 16;
    const int lane = threadIdx.x;
    const int half = lane >> 4;      // 0: K=k0,k0+1  1: K=k0+2,k0+3
    const int l    = lane & 15;

    v8f acc = {};
    for (int k0 = 0; k0 < D_; k0 += 4) {
        const int kA = k0 + half * 2;
        // A fragment: desc row n0+l, cols kA, kA+1
        const float* ap = desc + (size_t)(n0 + l) * D_ + kA;
        v2f a; a.x = ap[0]; a.y = ap[1];
        // B fragment: Wq rows e=kA (vgpr0) / kA+1 (vgpr1), col d0+l
        v2f b;
        b.x = Wq[(size_t)kA * D_ + d0 + l];
        b.y = Wq[(size_t)(kA + 1) * D_ + d0 + l];
        acc = __builtin_amdgcn_wmma_f32_16x16x4_f32(false, a, false, b,
                                                    (short)0, acc, false, false);
    }
    // C layout: vgpr r, lane l(+half*16)  ->  row n0 + r + half*8, col d0 + l
#pragma unroll
    for (int r = 0; r < 8; ++r) {
        Mt[(size_t)(d0 + l) * N_ + (n0 + r + half * 8)] = acc[r];
    }
}

// ---------------------------------------------------------------------------
// Kernel 1b: sb[n] = dot(bq, desc[n])  (bias folded into fused scores)
// ---------------------------------------------------------------------------
__global__ __launch_bounds__(32)
void bias_router_kernel(const float* __restrict__ bq,
                        const float* __restrict__ desc,
                        float* __restrict__ sb) {
    const int n = blockIdx.x;
    const int lane = threadIdx.x;
    float s = 0.f;
    for (int e = lane; e < D_; e += 32) s += bq[e] * desc[(size_t)n * D_ + e];
#pragma unroll
    for (int off = 16; off; off >>= 1) s += __shfl_down(s, off, 32);
    if (lane == 0) sb[n] = s;
}

// ---------------------------------------------------------------------------
// Kernel 2: fused scores-GEMM + mask + softmax + topk + sparse gather.
// Block = 16 wave32s (one WGP-sized workgroup).
// Phase 0: stage the contiguous 16x1024 f32 token tile into LDS with
//          GLOBAL_LOAD_ASYNC_TO_LDS_B128 (ASYNCcnt), bank-conflict padded.
// Phase 1: wave j computes the 16x16 score tile for set-columns 16j..16j+15
//          over K=D via V_WMMA_F32_16X16X4_F32 (A from LDS, B from L2-hot Mt).
// Phase 2: wave t owns token t: 8-sparse softmax / argmax / topk / gather.
// ---------------------------------------------------------------------------
__global__ __launch_bounds__(512)
void router_main_kernel(const float* __restrict__ tok,
                        const float* __restrict__ sets,
                        const float* __restrict__ Mt,
                        const float* __restrict__ sb,
                        const int*   __restrict__ t2s,
                        const float* __restrict__ temp_p,
                        float* __restrict__ out_repr,
                        int*   __restrict__ out_bank,
                        float* __restrict__ out_w,
                        int*   __restrict__ out_topk) {
    __shared__ __align__(16) float tokLds[16][D_ + TPAD];  // 64.25KB, padded
    __shared__ float sm[16][N_ + SPAD];                    // 16.25KB, padded

    const int wave = threadIdx.x >> 5;   // 0..15
    const int lane = threadIdx.x & 31;
    const int half = lane >> 4;
    const int l    = lane & 15;
    const int s0   = blockIdx.x * 16;
    const int b    = blockIdx.y;

    // ---- Phase 0: async GMEM->LDS stage of the token tile (64KB, contiguous)
    {
        const float* gbase = tok + ((size_t)(b * S_ + s0)) * D_;
        const int tid = threadIdx.x;
#pragma unroll
        for (int i = 0; i < (16 * D_) / (512 * 4); ++i) {   // 8 x b128 per thread
            const int flat = (i * 512 + tid) * 4;           // 4 floats / chunk
            const int r = flat / D_;
            const int c = flat % D_;                         // multiple of 4 -> 16B aligned
            uint32_t lds_off = (uint32_t)(uintptr_t)&tokLds[r][c];
            unsigned long long gaddr = (unsigned long long)(uintptr_t)(gbase + flat);
            asm volatile("global_load_async_to_lds_b128 %0, %1, off"
                         :: "v"(lds_off), "v"(gaddr)
                         : "memory");
        }
        asm volatile("s_wait_asynccnt 0x0" ::: "memory");
    }
    __syncthreads();

    // ---- Phase 1: scores tile via V_WMMA_F32_16X16X4_F32 ----
    {
        const int n0 = wave * 16;
        v8f acc = {};
        const float* arow = &tokLds[l][0];   // LDS, conflict-free b64 reads
        for (int kb = 0; kb < D_; kb += 64) {
            // pull the next B-operand block toward the caches (global_prefetch_b8)
            const int kpf = (kb + 64 < D_) ? kb + 64 : kb;
            __builtin_prefetch(Mt + (size_t)kpf * N_ + n0 + l, 0, 1);
#pragma unroll
            for (int k0 = kb; k0 < kb + 64; k0 += 4) {
                const int kA = k0 + half * 2;
                v2f a; a.x = arow[kA]; a.y = arow[kA + 1];
                v2f bb;
                bb.x = Mt[(size_t)kA * N_ + n0 + l];        // coalesced across lanes
                bb.y = Mt[(size_t)(kA + 1) * N_ + n0 + l];
                acc = __builtin_amdgcn_wmma_f32_16x16x4_f32(false, a, false, bb,
                                                            (short)0, acc, false, false);
            }
        }
#pragma unroll
        for (int r = 0; r < 8; ++r) {
            const int m = r + half * 8;
            const int n = n0 + l;
            sm[m][n] = (acc[r] + sb[n]) * SCALE;
        }
    }
    __syncthreads();

    // ---- Phase 2: wave `wave` handles token s0+wave ----
    const int t = wave;
    const int s = s0 + t;
    const float temp = fmaxf(temp_p[0], MIN_T);

    // lanes 0..7 fetch routing candidates
    int colk = 0, validk = 0; float sck = NEGV;
    if (lane < K_) {
        const int idx = t2s[(size_t)s * K_ + lane];
        validk = (idx >= 0);
        colk   = min(max(idx, 0), N_ - 1);
        sck    = sm[t][colk];
    }
    // broadcast the 8-entry table into every lane (wave-uniform after this)
    int cols[K_], vld[K_], own[K_]; float scv[K_];
#pragma unroll
    for (int j = 0; j < K_; ++j) {
        cols[j] = __shfl(colk, j, 32);
        vld[j]  = __shfl(validk, j, 32);
        scv[j]  = __shfl(sck, j, 32);
    }
    // dedup: entry owns its column iff valid and no earlier valid entry has it
#pragma unroll
    for (int j = 0; j < K_; ++j) {
        int o = vld[j];
        for (int i = 0; i < j; ++i) o = o && !(vld[i] && cols[i] == cols[j]);
        own[j] = o;
    }

    // softmax over the (<=8) live columns; NEG columns underflow to exactly 0
    float mx = NEGV;
#pragma unroll
    for (int j = 0; j < K_; ++j) if (own[j] && scv[j] > mx) mx = scv[j];
    float Z = 0.f;
#pragma unroll
    for (int j = 0; j < K_; ++j) if (own[j]) Z += expf((scv[j] - mx) / temp);
    const float invZ = 1.f / Z;
    float wv[K_];
#pragma unroll
    for (int j = 0; j < K_; ++j)
        wv[j] = own[j] ? expf((scv[j] - mx) / temp) * invZ : 0.f;

    // bank_indices = argmax(weights): highest score, tie -> lowest column
    int bank = 0; float bs = NEGV;
#pragma unroll
    for (int j = 0; j < K_; ++j)
        if (own[j] && (scv[j] > bs || (scv[j] == bs && cols[j] < bank))) {
            bs = scv[j]; bank = cols[j];
        }
    if (lane == 0) out_bank[(size_t)b * S_ + s] = bank;

    // topk_idx: owners sorted by (score desc, col asc); remaining slots take
    // the lowest non-member columns (the NEG ties resolve to lowest index).
    if (lane == 0) {
        int m = 0;
#pragma unroll
        for (int j = 0; j < K_; ++j) m += own[j];
        int tk[K_];
#pragma unroll
        for (int k = 0; k < K_; ++k) {
            if (own[k]) {
                int rank = 0;
                for (int j = 0; j < K_; ++j)
                    if (own[j] && (scv[j] > scv[k] ||
                                   (scv[j] == scv[k] && cols[j] < cols[k]))) rank++;
                tk[rank] = cols[k];
            }
        }
        int p = m;
        for (int c = 0; c < 2 * K_ && p < K_; ++c) {   // owners<=8 -> c<16 suffices
            bool member = false;
            for (int j = 0; j < K_; ++j) member = member || (own[j] && cols[j] == c);
            if (!member) tk[p++] = c;
        }
        int* tp = out_topk + ((size_t)(b * S_ + s)) * K_;
#pragma unroll
        for (int k = 0; k < K_; ++k) tp[k] = tk[k];
    }

    // weights row [N_]: 8 columns per lane, sparse fill
    {
        float* wp = out_w + ((size_t)(b * S_ + s)) * N_;
#pragma unroll
        for (int u = 0; u < N_ / 32; ++u) {
            const int c = lane * (N_ / 32) + u;
            float w = 0.f;
#pragma unroll
            for (int j = 0; j < K_; ++j) if (own[j] && cols[j] == c) w = wv[j];
            wp[c] = w;
        }
    }

    // token_repr[b,s,:] = sum_j wv[j] * set_states[b, cols[j], :]  (8-sparse)
    {
        float accr[D_ / 32];
#pragma unroll
        for (int i = 0; i < D_ / 32; ++i) accr[i] = 0.f;
#pragma unroll
        for (int j = 0; j < K_; ++j) {
            if (own[j]) {
                const float w = wv[j];
                const float* sp = sets + ((size_t)b * N_ + cols[j]) * D_ + lane;
#pragma unroll
                for (int i = 0; i < D_ / 32; ++i) accr[i] += w * sp[i * 32];
            }
        }
        float* op = out_repr + ((size_t)(b * S_ + s)) * D_ + lane;
#pragma unroll
        for (int i = 0; i < D_ / 32; ++i) op[i * 32] = accr[i];
    }
}

// ---------------------------------------------------------------------------
extern "C" void kernel_launch(void* const* d_in, const int* in_sizes, int n_in,
                              void* d_out, int out_size, void* d_ws, size_t ws_size,
                              hipStream_t stream) {
    const float* tok  = (const float*)d_in[0];   // [B,S,D]
    const float* sets = (const float*)d_in[1];   // [B,N,D]
    const float* desc = (const float*)d_in[2];   // [N,D]
    const int*   t2s  = (const int*)  d_in[3];   // [S,K]
    const float* Wq   = (const float*)d_in[4];   // [D,D]
    const float* bq   = (const float*)d_in[5];   // [D]
    const float* temp = (const float*)d_in[6];   // [1]

    // outputs concatenated flat: token_repr, bank_indices, weights, topk_idx
    float* out      = (float*)d_out;
    float* out_repr = out;                                            // B*S*D
    int*   out_bank = (int*)(out + (size_t)B_ * S_ * D_);             // B*S
    float* out_w    = out + (size_t)B_ * S_ * D_ + (size_t)B_ * S_;   // B*S*N
    int*   out_topk = (int*)(out_w + (size_t)B_ * S_ * N_);           // B*S*K

    // workspace: Mt [D][N] (1MB) + sb [N]
    float* Mt = (float*)d_ws;
    float* sb = Mt + (size_t)D_ * N_;

    dim3 g1(D_ / 16, N_ / 16);
    fuse_router_kernel<<<g1, 32, 0, stream>>>(desc, Wq, Mt);
    bias_router_kernel<<<N_, 32, 0, stream>>>(bq, desc, sb);

    dim3 g2(S_ / 16, B_);
    router_main_kernel<<<g2, 512, 0, stream>>>(tok, sets, Mt, sb, t2s, temp,
                                               out_repr, out_bank, out_w, out_topk);
}